// SAGE_16209206575326
// MI455X (gfx1250) — compile-verified
//
#include <hip/hip_runtime.h>
#include <hip/hip_bf16.h>
#include <math.h>

typedef float v2f __attribute__((ext_vector_type(2)));
typedef float v8f __attribute__((ext_vector_type(8)));

#define NN      50000
#define NE      600000
#define INC     128
#define HIDC    128
#define OUTC    64
#define NG      512

__device__ __forceinline__ void atomAddF(float* p, float v) {
  __hip_atomic_fetch_add(p, v, __ATOMIC_RELAXED, __HIP_MEMORY_SCOPE_AGENT);
}

// ---------------- degree ----------------
__global__ void deg_kernel(const int* __restrict__ dst, float* __restrict__ deg, int E) {
  int e = blockIdx.x * blockDim.x + threadIdx.x;
  if (e < E) atomAddF(&deg[dst[e]], 1.0f);
}

__global__ void invdeg_kernel(float* __restrict__ deg, int n) {
  int i = blockIdx.x * blockDim.x + threadIdx.x;
  if (i < n) deg[i] = 1.0f / fmaxf(deg[i], 1.0f);
}

// ---------------- edge aggregation: agg[dst] += x[src], 128 ch ----------------
// one wave per edge; lane handles 4 consecutive channels (float4 load, 4 atomics)
__global__ void edge_agg_kernel(const float* __restrict__ x, const int* __restrict__ src,
                                const int* __restrict__ dst, float* __restrict__ agg, int E) {
  int wave = blockIdx.x * (blockDim.x >> 5) + (threadIdx.x >> 5);
  int lane = threadIdx.x & 31;
  if (wave >= E) return;
  int s = src[wave];
  int d = dst[wave];
  const float4 v = *reinterpret_cast<const float4*>(x + (size_t)s * 128 + lane * 4);
  float* p = agg + (size_t)d * 128 + lane * 4;
  atomAddF(p + 0, v.x);
  atomAddF(p + 1, v.y);
  atomAddF(p + 2, v.z);
  atomAddF(p + 3, v.w);
}

// ---------------- fused SAGE GEMM via V_WMMA_F32_16X16X4_F32 ----------------
// Out = (Agg * invdeg) @ Wl + bl + X @ Wr   ; per-wave 16x16 tile, K-loop step 4.
// A frag (16x4 f32): lanes 0-15 -> M=lane, K={0,1}; lanes 16-31 -> M=lane-16, K={2,3}
// B frag (4x16 f32): lanes 0-15 -> N=lane, K={0,1}; lanes 16-31 -> N=lane-16, K={2,3}
// C/D (16x16 f32, 8 VGPR): vgpr r, lanes 0-15 -> M=r, N=lane; lanes 16-31 -> M=r+8
__global__ void sage_gemm_kernel(const float* __restrict__ Agg, const float* __restrict__ invdeg,
                                 const float* __restrict__ X,
                                 const float* __restrict__ Wl, const float* __restrict__ bl,
                                 const float* __restrict__ Wr,
                                 float* __restrict__ OutA, float* __restrict__ OutB,
                                 int M, int K, int N, int doRelu) {
  int wave = blockIdx.x * (blockDim.x >> 5) + (threadIdx.x >> 5);
  int lane = threadIdx.x & 31;
  int nTilesN = N >> 4;
  int nTiles = (M >> 4) * nTilesN;
  if (wave >= nTiles) return;          // wave-uniform: EXEC stays all-ones for WMMA

  int tn = wave % nTilesN;
  int tm = wave / nTilesN;
  int m0 = tm << 4;
  int n0 = tn << 4;

  int rA   = m0 + (lane & 15);         // A-fragment row this lane owns
  int colB = n0 + (lane & 15);         // B-fragment column this lane owns
  int kh   = (lane >> 4) << 1;         // K sub-offset: 0 (lo half) or 2 (hi half)

  const float* arow = Agg + (size_t)rA * K;
  const float* xrow = X   + (size_t)rA * K;
  float s = invdeg[rA];

  v8f c = {};
#pragma unroll 4
  for (int kk = 0; kk < K; kk += 4) {
    v2f a, b;
    // mean-aggregated neighbors @ Wl
    a.x = arow[kk + kh] * s;
    a.y = arow[kk + kh + 1] * s;
    b.x = Wl[(size_t)(kk + kh) * N + colB];
    b.y = Wl[(size_t)(kk + kh + 1) * N + colB];
    c = __builtin_amdgcn_wmma_f32_16x16x4_f32(false, a, false, b, (short)0, c, false, false);
    // self features @ Wr
    a.x = xrow[kk + kh];
    a.y = xrow[kk + kh + 1];
    b.x = Wr[(size_t)(kk + kh) * N + colB];
    b.y = Wr[(size_t)(kk + kh + 1) * N + colB];
    c = __builtin_amdgcn_wmma_f32_16x16x4_f32(false, a, false, b, (short)0, c, false, false);
  }

  float bias = bl[colB];
  int rbase = m0 + ((lane >> 4) << 3);
#pragma unroll
  for (int r = 0; r < 8; ++r) {
    float v = c[r] + bias;
    size_t off = (size_t)(rbase + r) * N + colB;
    if (OutB) OutB[off] = v;           // pre-ReLU copy (layer-1 "out")
    if (doRelu) v = fmaxf(v, 0.0f);
    OutA[off] = v;
  }
}

// ---------------- graph pooling: gsum[cluster[n]] += h[n]; gcnt[cluster[n]]++ ----------------
__global__ void pool_kernel(const float* __restrict__ h, const int* __restrict__ cluster,
                            float* __restrict__ gsum, float* __restrict__ gcnt, int n) {
  int idx = blockIdx.x * blockDim.x + threadIdx.x;
  int node = idx >> 5;
  int lane = idx & 31;
  if (node >= n) return;
  int g = cluster[node];
  const float4 v = *reinterpret_cast<const float4*>(h + (size_t)node * 128 + lane * 4);
  float* p = gsum + (size_t)g * 128 + lane * 4;
  atomAddF(p + 0, v.x);
  atomAddF(p + 1, v.y);
  atomAddF(p + 2, v.z);
  atomAddF(p + 3, v.w);
  if (lane == 0) atomAddF(&gcnt[g], 1.0f);
}

__global__ void pool_div_kernel(float* __restrict__ g, const float* __restrict__ gcnt, int total) {
  int i = blockIdx.x * blockDim.x + threadIdx.x;
  if (i >= total) return;
  g[i] = g[i] / fmaxf(gcnt[i >> 7], 1.0f);
}

// ---------------- in-place log-softmax over rows of 64 (one wave32 per row) ----------------
__global__ void log_softmax_kernel(float* __restrict__ out, int rows) {
  int wave = blockIdx.x * (blockDim.x >> 5) + (threadIdx.x >> 5);
  int lane = threadIdx.x & 31;
  if (wave >= rows) return;
  float* row = out + (size_t)wave * 64;
  float v0 = row[lane];
  float v1 = row[lane + 32];
  float m = fmaxf(v0, v1);
#pragma unroll
  for (int off = 16; off >= 1; off >>= 1) m = fmaxf(m, __shfl_xor(m, off, 32));
  float s = __expf(v0 - m) + __expf(v1 - m);
#pragma unroll
  for (int off = 16; off >= 1; off >>= 1) s += __shfl_xor(s, off, 32);
  float lse = m + __logf(s);
  row[lane]      = v0 - lse;
  row[lane + 32] = v1 - lse;
}

extern "C" void kernel_launch(void* const* d_in, const int* in_sizes, int n_in,
                              void* d_out, int out_size, void* d_ws, size_t ws_size,
                              hipStream_t stream) {
  const float* x       = (const float*)d_in[0];
  const int*   ei      = (const int*)d_in[1];
  const int*   src     = ei;
  const int*   dst     = ei + NE;
  const int*   cluster = (const int*)d_in[2];
  const float* Wl0 = (const float*)d_in[3];
  const float* bl0 = (const float*)d_in[4];
  const float* Wr0 = (const float*)d_in[5];
  const float* Wl1 = (const float*)d_in[6];
  const float* bl1 = (const float*)d_in[7];
  const float* Wr1 = (const float*)d_in[8];
  const float* Wl2 = (const float*)d_in[9];
  const float* bl2 = (const float*)d_in[10];
  const float* Wr2 = (const float*)d_in[11];

  // d_out layout: [log_softmax 50000*64][pre-relu h2 50000*128][g 512*128]
  float* out_ls = (float*)d_out;
  float* out_h2 = out_ls + (size_t)NN * OUTC;
  float* out_g  = out_h2 + (size_t)NN * HIDC;

  float* ws   = (float*)d_ws;
  float* agg  = ws;                               // 6,400,000 f
  float* h1   = ws + (size_t)NN * HIDC;           // 6,400,000 f
  float* h2r  = h1 + (size_t)NN * HIDC;           // 6,400,000 f
  float* deg  = h2r + (size_t)NN * HIDC;          // 50,000 f  (becomes invdeg)
  float* gcnt = deg + NN;                         // 512 f

  const int TPB = 256;
  const int edgeWaveBlocks = (NE + 7) / 8;        // 8 waves/block, 1 wave/edge

  // degrees -> invdeg
  hipMemsetAsync(deg, 0, NN * sizeof(float), stream);
  deg_kernel<<<(NE + TPB - 1) / TPB, TPB, 0, stream>>>(dst, deg, NE);
  invdeg_kernel<<<(NN + TPB - 1) / TPB, TPB, 0, stream>>>(deg, NN);

  // ---- layer 0: h1 = relu(mean(x[src]->dst) @ Wl0 + bl0 + x @ Wr0)
  hipMemsetAsync(agg, 0, (size_t)NN * HIDC * sizeof(float), stream);
  edge_agg_kernel<<<edgeWaveBlocks, TPB, 0, stream>>>(x, src, dst, agg, NE);
  {
    int tiles = (NN / 16) * (HIDC / 16);
    sage_gemm_kernel<<<(tiles + 7) / 8, TPB, 0, stream>>>(
        agg, deg, x, Wl0, bl0, Wr0, h1, nullptr, NN, INC, HIDC, 1);
  }

  // ---- layer 1: out = mean(h1)@Wl1 + bl1 + h1@Wr1 ; h2r = relu(out)
  hipMemsetAsync(agg, 0, (size_t)NN * HIDC * sizeof(float), stream);
  edge_agg_kernel<<<edgeWaveBlocks, TPB, 0, stream>>>(h1, src, dst, agg, NE);
  {
    int tiles = (NN / 16) * (HIDC / 16);
    sage_gemm_kernel<<<(tiles + 7) / 8, TPB, 0, stream>>>(
        agg, deg, h1, Wl1, bl1, Wr1, h2r, out_h2, NN, HIDC, HIDC, 1);
  }

  // ---- graph pooling on h2r -> out_g
  hipMemsetAsync(out_g, 0, (size_t)NG * HIDC * sizeof(float), stream);
  hipMemsetAsync(gcnt, 0, NG * sizeof(float), stream);
  pool_kernel<<<((size_t)NN * 32 + TPB - 1) / TPB, TPB, 0, stream>>>(h2r, cluster, out_g, gcnt, NN);
  pool_div_kernel<<<(NG * HIDC + TPB - 1) / TPB, TPB, 0, stream>>>(out_g, gcnt, NG * HIDC);

  // ---- layer 2: xf = mean(h2r)@Wl2 + bl2 + h2r@Wr2 ; then log_softmax in place
  hipMemsetAsync(agg, 0, (size_t)NN * HIDC * sizeof(float), stream);
  edge_agg_kernel<<<edgeWaveBlocks, TPB, 0, stream>>>(h2r, src, dst, agg, NE);
  {
    int tiles = (NN / 16) * (OUTC / 16);
    sage_gemm_kernel<<<(tiles + 7) / 8, TPB, 0, stream>>>(
        agg, deg, h2r, Wl2, bl2, Wr2, out_ls, nullptr, NN, HIDC, OUTC, 0);
  }
  log_softmax_kernel<<<((size_t)NN * 32 + TPB - 1) / TPB, TPB, 0, stream>>>(out_ls, NN);
}